// NonlinearIB_73169062854905
// MI455X (gfx1250) — compile-verified
//
#include <hip/hip_runtime.h>

#define N_PTS 8192
#define DIMS  512
#define BM 64
#define BN 64
#define SD (DIMS + 4)                 // padded LDS row stride (floats): 4-bank skew
#define COLSPLIT 2
#define CTILES ((N_PTS / BN) / COLSPLIT)   // 64 column tiles per split

typedef __attribute__((ext_vector_type(2))) float v2f;
typedef __attribute__((ext_vector_type(4))) float v4f;
typedef __attribute__((ext_vector_type(8))) float v8f;

#define LOG2E 1.44269504088896340736f
#define LN2   0.69314718055994530942f

static __device__ __forceinline__ void lse_merge(float& m1, float& s1, float m2, float s2) {
  float m = fmaxf(m1, m2);
  s1 = s1 * exp2f(m1 - m) + s2 * exp2f(m2 - m);
  m1 = m;
}

// ---------------------------------------------------------------------------
// Kernel 1: y = x + exp(0.5*noise_logvar)*noise ; x2[row] = sum(x[row]^2)
// ---------------------------------------------------------------------------
__global__ __launch_bounds__(256) void prep_kernel(const float* __restrict__ x,
                                                   const float* __restrict__ noise,
                                                   const float* __restrict__ nlv_p,
                                                   float* __restrict__ y,
                                                   float* __restrict__ x2) {
  const int row  = blockIdx.x;
  const int tid  = threadIdx.x;
  const int lane = tid & 31;
  const int wv   = tid >> 5;
  const float sigma = exp2f(0.5f * LOG2E * nlv_p[0]);   // exp(0.5*nlv)

  float acc = 0.f;
  #pragma unroll
  for (int c = tid; c < DIMS; c += 256) {
    float v = x[row * DIMS + c];
    y[row * DIMS + c] = fmaf(sigma, noise[row * DIMS + c], v);
    acc = fmaf(v, v, acc);
  }
  #pragma unroll
  for (int off = 16; off > 0; off >>= 1) acc += __shfl_xor(acc, off, 32);

  __shared__ float wsum[8];
  if (lane == 0) wsum[wv] = acc;
  __syncthreads();
  if (tid == 0) {
    float t = 0.f;
    #pragma unroll
    for (int i = 0; i < 8; ++i) t += wsum[i];
    x2[row] = t;
  }
}

// ---------------------------------------------------------------------------
// Kernel 2: fused Gram (f32 WMMA) + dual online logsumexp over columns.
//   variant A: scores -dists/(2*total_var), diagonal included (max == 0)
//   variant B: scores -dists/(2*kde_var),  diagonal masked, online (max,sum)
// All score math done in the exp2/log2 domain (v_exp_f32 is 2^x).
// ---------------------------------------------------------------------------
__global__ __launch_bounds__(256, 1)
void gram_lse_kernel(const float* __restrict__ x, const float* __restrict__ x2g,
                     const float* __restrict__ nlv_p, const float* __restrict__ klv_p,
                     float* __restrict__ sumA_part, float* __restrict__ mB_part,
                     float* __restrict__ sB_part) {
  extern __shared__ float lds[];
  float* aBuf = lds;               // BM * SD floats (resident A rows)
  float* bBuf = lds + BM * SD;     // BN * SD floats (streamed B tiles)

  const int tid   = threadIdx.x;
  const int wv    = tid >> 5;
  const int lane  = tid & 31;
  const int hh    = lane >> 4;     // lane half (selects K sub-pair / M|N sub-block)
  const int ln16  = lane & 15;
  const int rowBlock = blockIdx.x * BM;
  const int split    = blockIdx.y;

  const float nlv = nlv_p[0], klv = klv_p[0];
  const float tvA  = exp2f(nlv * LOG2E) + exp2f(klv * LOG2E);  // total_var
  const float kvar = exp2f(klv * LOG2E);                        // kde var
  // score2 = g*iX - q*hX   (log2 domain), q = x2_row + x2_col
  const float iA = LOG2E / tvA,  hA = 0.5f * LOG2E / tvA;
  const float iB = LOG2E / kvar, hB = 0.5f * LOG2E / kvar;

  // stage the 64 A rows into LDS once (coalesced b128 loads)
  for (int v = tid; v < BM * (DIMS / 4); v += 256) {
    int r = v >> 7, c4 = v & 127;
    v4f d = *(const v4f*)(x + (rowBlock + r) * DIMS + c4 * 4);
    *(v4f*)(aBuf + r * SD + c4 * 4) = d;
  }

  const int rsub  = wv & 3;              // row sub-tile (16 rows)
  const int csub0 = (wv >> 2) << 1;      // two column sub-tiles: csub0, csub0+1

  float x2r[8];
  #pragma unroll
  for (int r = 0; r < 8; ++r) x2r[r] = x2g[rowBlock + rsub * 16 + hh * 8 + r];

  float sumA[2][8], mB[2][8], sB[2][8];
  #pragma unroll
  for (int a = 0; a < 2; ++a)
    #pragma unroll
    for (int r = 0; r < 8; ++r) { sumA[a][r] = 0.f; mB[a][r] = -3.0e38f; sB[a][r] = 0.f; }

  const float* aRow  = aBuf + (rsub * 16 + ln16) * SD + hh * 2;
  const float* bRow0 = bBuf + (csub0 * 16 + ln16) * SD + hh * 2;
  const float* bRow1 = bRow0 + 16 * SD;

  for (int ct = split * CTILES; ct < (split + 1) * CTILES; ++ct) {
    const int colBlock = ct * BN;
    __syncthreads();
    for (int v = tid; v < BN * (DIMS / 4); v += 256) {
      int r = v >> 7, c4 = v & 127;
      v4f d = *(const v4f*)(x + (colBlock + r) * DIMS + c4 * 4);
      *(v4f*)(bBuf + r * SD + c4 * 4) = d;
    }
    __syncthreads();

    v8f c0 = {}; v8f c1 = {};
    #pragma unroll 16
    for (int k = 0; k < DIMS; k += 4) {
      v2f a  = *(const v2f*)(aRow  + k);
      v2f b0 = *(const v2f*)(bRow0 + k);
      v2f b1 = *(const v2f*)(bRow1 + k);
      c0 = __builtin_amdgcn_wmma_f32_16x16x4_f32(false, a, false, b0, (short)0, c0, false, false);
      c1 = __builtin_amdgcn_wmma_f32_16x16x4_f32(false, a, false, b1, (short)0, c1, false, false);
    }

    const float colT0 = x2g[colBlock + csub0 * 16 + ln16];
    const float colT1 = x2g[colBlock + (csub0 + 1) * 16 + ln16];
    const bool diagTile = (colBlock == rowBlock);
    const int rloc = rsub * 16 + hh * 8;   // + r = local row in 64-block

    #pragma unroll
    for (int r = 0; r < 8; ++r) {
      {  // accumulator 0
        float g = c0[r];
        float q = x2r[r] + colT0;
        sumA[0][r] += exp2f(fmaf(g, iA, -q * hA));
        float s = fmaf(g, iB, -q * hB);
        bool msk = diagTile && ((rloc + r) == (csub0 * 16 + ln16));
        if (msk) s = -3.0e38f;
        float m2 = fmaxf(mB[0][r], s);
        float e  = msk ? 0.f : exp2f(s - m2);
        sB[0][r] = sB[0][r] * exp2f(mB[0][r] - m2) + e;
        mB[0][r] = m2;
      }
      {  // accumulator 1
        float g = c1[r];
        float q = x2r[r] + colT1;
        sumA[1][r] += exp2f(fmaf(g, iA, -q * hA));
        float s = fmaf(g, iB, -q * hB);
        bool msk = diagTile && ((rloc + r) == ((csub0 + 1) * 16 + ln16));
        if (msk) s = -3.0e38f;
        float m2 = fmaxf(mB[1][r], s);
        float e  = msk ? 0.f : exp2f(s - m2);
        sB[1][r] = sB[1][r] * exp2f(mB[1][r] - m2) + e;
        mB[1][r] = m2;
      }
    }
  }

  // --- reduce: merge the two column sub-tiles, then 16-lane butterfly ---
  float mW[8], sW[8], aW[8];
  #pragma unroll
  for (int r = 0; r < 8; ++r) {
    float m = mB[0][r], s = sB[0][r];
    lse_merge(m, s, mB[1][r], sB[1][r]);
    float sa = sumA[0][r] + sumA[1][r];
    #pragma unroll
    for (int off = 1; off < 16; off <<= 1) {
      float mo = __shfl_xor(m,  off, 32);
      float so = __shfl_xor(s,  off, 32);
      float ao = __shfl_xor(sa, off, 32);
      lse_merge(m, s, mo, so);
      sa += ao;
    }
    mW[r] = m; sW[r] = s; aW[r] = sa;
  }

  // --- merge the two waves covering each row (via LDS), emit partials ---
  __syncthreads();
  float* smM = lds;        // 128 entries (wave,localrow)
  float* smS = lds + 128;
  float* smA = lds + 256;
  if (ln16 == 0) {
    #pragma unroll
    for (int r = 0; r < 8; ++r) {
      int idx = wv * 16 + hh * 8 + r;
      smM[idx] = mW[r]; smS[idx] = sW[r]; smA[idx] = aW[r];
    }
  }
  __syncthreads();
  if (tid < BM) {
    int row = tid;
    int i0  = (row >> 4) * 16 + (row & 15);
    int i1  = i0 + 4 * 16;
    float m = smM[i0], s = smS[i0];
    lse_merge(m, s, smM[i1], smS[i1]);
    float sa = smA[i0] + smA[i1];
    int o = split * N_PTS + rowBlock + row;
    sumA_part[o] = sa;
    mB_part[o]   = m;
    sB_part[o]   = s;
  }
}

// ---------------------------------------------------------------------------
// Kernel 3: combine column-split partials, compute the two scalar losses.
// ---------------------------------------------------------------------------
__global__ __launch_bounds__(256) void finalize_kernel(
    const float* __restrict__ sumA_part, const float* __restrict__ mB_part,
    const float* __restrict__ sB_part,
    const float* __restrict__ nlv_p, const float* __restrict__ klv_p,
    const float* __restrict__ beta_p, float* __restrict__ out2) {
  const int tid  = threadIdx.x;
  const int lane = tid & 31;
  const int wv   = tid >> 5;

  float accA = 0.f, accB = 0.f;
  for (int row = tid; row < N_PTS; row += 256) {
    float m = mB_part[row], s = sB_part[row];
    lse_merge(m, s, mB_part[N_PTS + row], sB_part[N_PTS + row]);
    float sa = sumA_part[row] + sumA_part[N_PTS + row];
    accA += log2f(sa) * LN2;               // natural-log LSE, variant A
    accB += (m + log2f(s)) * LN2;          // natural-log LSE, variant B
  }
  #pragma unroll
  for (int off = 16; off > 0; off >>= 1) {
    accA += __shfl_xor(accA, off, 32);
    accB += __shfl_xor(accB, off, 32);
  }
  __shared__ float rA[8], rB[8];
  if (lane == 0) { rA[wv] = accA; rB[wv] = accB; }
  __syncthreads();
  if (tid == 0) {
    float tA = 0.f, tB = 0.f;
    #pragma unroll
    for (int i = 0; i < 8; ++i) { tA += rA[i]; tB += rB[i]; }
    const float meanA = tA / (float)N_PTS;
    const float meanB = tB / (float)N_PTS;
    const float nlv = nlv_p[0], klv = klv_p[0], beta = beta_p[0];
    const float two_pi = 6.283185307179586f;

    const float tv    = expf(nlv) + expf(klv);
    const float normA = (DIMS * 0.5f) * logf(two_pi * tv);
    const float lpA   = meanA - logf((float)N_PTS) - normA;  // mean lprobs
    const float h     = -lpA;
    const float mi    = h - normA;
    out2[0] = beta * mi;                                     // ib_loss

    const float kvar  = expf(klv);
    const float normK = (DIMS * 0.5f) * logf(two_pi * kvar);
    const float lpB   = meanB - logf((float)(N_PTS - 1)) - normK;
    out2[1] = -lpB;                                          // kde_loo_loss
  }
}

// ---------------------------------------------------------------------------
extern "C" void kernel_launch(void* const* d_in, const int* in_sizes, int n_in,
                              void* d_out, int out_size, void* d_ws, size_t ws_size,
                              hipStream_t stream) {
  const float* x     = (const float*)d_in[0];
  const float* noise = (const float*)d_in[1];
  const float* nlv   = (const float*)d_in[2];
  const float* klv   = (const float*)d_in[3];
  const float* beta  = (const float*)d_in[4];

  float* y    = (float*)d_out;
  float* out2 = (float*)d_out + (size_t)N_PTS * DIMS;

  float* x2        = (float*)d_ws;              // N
  float* sumA_part = x2 + N_PTS;                // 2*N
  float* mB_part   = sumA_part + 2 * N_PTS;     // 2*N
  float* sB_part   = mB_part + 2 * N_PTS;       // 2*N

  prep_kernel<<<N_PTS, 256, 0, stream>>>(x, noise, nlv, y, x2);

  const int ldsBytes = (BM + BN) * SD * (int)sizeof(float);   // 258 KB < 320 KB/WGP
  hipFuncSetAttribute(reinterpret_cast<const void*>(gram_lse_kernel),
                      hipFuncAttributeMaxDynamicSharedMemorySize, ldsBytes);
  dim3 grid(N_PTS / BM, COLSPLIT);
  gram_lse_kernel<<<grid, 256, ldsBytes, stream>>>(x, x2, nlv, klv,
                                                   sumA_part, mB_part, sB_part);

  finalize_kernel<<<1, 256, 0, stream>>>(sumA_part, mB_part, sB_part,
                                         nlv, klv, beta, out2);
}